// cp_39582418600045
// MI455X (gfx1250) — compile-verified
//
#include <hip/hip_runtime.h>

// CP-model sampled evaluation for MI455X (gfx1250, wave32).
// out[n] = sum_r lamb[r] * f0[i0[n],r] * f1[i1[n],r] * f2[i2[n],r],  R = 64.
//
// One wave handles 16 entries. The r-reduction is done on the matrix pipe:
//   A[m][k] = lamb[k] * f0[i0[m],k] * f1[i1[m],k] * f2[i2[m],k]   (16x4 chunk)
//   B       = all-ones 4x16
//   D      += A x B   -> every column of D equals the desired 16 outputs.
// 16 accumulating V_WMMA_F32_16X16X4_F32 ops sweep K = 0..63.

typedef __attribute__((ext_vector_type(2))) float v2f;
typedef __attribute__((ext_vector_type(8))) float v8f;

#define RANK 64

__global__ __launch_bounds__(256) void cp_eval_wmma_kernel(
    const int*   __restrict__ idx,   // [N,3] int32
    const float* __restrict__ f0,    // [S,64]
    const float* __restrict__ f1,    // [S,64]
    const float* __restrict__ f2,    // [S,64]
    const float* __restrict__ lamb,  // [64]
    float*       __restrict__ out,   // [N]
    int n)
{
    const int lane  = threadIdx.x & 31;
    const int wave  = blockIdx.x * (blockDim.x >> 5) + (threadIdx.x >> 5);
    const int base  = wave * 16;                 // wave-uniform
    if (base >= n) return;                       // uniform exit: EXEC stays all-1s for WMMA

    const int m     = lane & 15;                 // entry (row M) this lane feeds
    const int khalf = lane >> 4;                 // 0 -> K={0,1}, 1 -> K={2,3} of each chunk
    const int e     = base + m;

    const int i0 = idx[3 * e + 0];
    const int i1 = idx[3 * e + 1];
    const int i2 = idx[3 * e + 2];

    const float* __restrict__ r0 = f0 + (size_t)i0 * RANK;
    const float* __restrict__ r1 = f1 + (size_t)i1 * RANK;
    const float* __restrict__ r2 = f2 + (size_t)i2 * RANK;

    v8f c = {};                                  // 16x16 f32 accumulator (8 VGPRs)
    v2f bones;                                   // B = ones: layout-independent
    bones.x = 1.0f;
    bones.y = 1.0f;

#pragma unroll
    for (int j = 0; j < 16; ++j) {
        const int r = 4 * j + 2 * khalf;         // this lane's K-pair within chunk j
        const float w0 = lamb[r];
        const float w1 = lamb[r + 1];
        v2f a;
        a.x = w0 * r0[r]     * r1[r]     * r2[r];       // A[m][chunk K + 0/2]
        a.y = w1 * r0[r + 1] * r1[r + 1] * r2[r + 1];   // A[m][chunk K + 1/3]
        // 8 args: (neg_a, A, neg_b, B, c_mod, C, reuse_a, reuse_b)
        c = __builtin_amdgcn_wmma_f32_16x16x4_f32(
                false, a, false, bones, (short)0, c, false, false);
    }

    // D layout (32-bit C/D 16x16): lane 0 holds D[0..7][col 0] in c[0..7],
    // lane 16 holds D[8..15][col 0]. All columns identical -> two lanes store
    // all 16 results as 128-bit writes.
    if ((lane & 15) == 0) {
        float4 lo = make_float4(c[0], c[1], c[2], c[3]);
        float4 hi = make_float4(c[4], c[5], c[6], c[7]);
        float4* o = reinterpret_cast<float4*>(out + base + 8 * khalf);
        o[0] = lo;   // rows {0..3} or {8..11}
        o[1] = hi;   // rows {4..7} or {12..15}
    }
}

extern "C" void kernel_launch(void* const* d_in, const int* in_sizes, int n_in,
                              void* d_out, int out_size, void* d_ws, size_t ws_size,
                              hipStream_t stream) {
    const int*   idx  = (const int*)d_in[0];    // [N,3]
    const float* f0   = (const float*)d_in[1];  // [S,64]
    const float* f1   = (const float*)d_in[2];
    const float* f2   = (const float*)d_in[3];
    const float* lamb = (const float*)d_in[4];  // [64]
    float*       out  = (float*)d_out;

    const int n = in_sizes[0] / 3;              // number of entries
    const int waves  = (n + 15) / 16;           // one wave per 16 entries
    const int wpb    = 256 / 32;                // 8 waves per block
    const int blocks = (waves + wpb - 1) / wpb;

    cp_eval_wmma_kernel<<<blocks, 256, 0, stream>>>(idx, f0, f1, f2, lamb, out, n);
}